// ChebNet_15444702396430
// MI455X (gfx1250) — compile-verified
//
#include <hip/hip_runtime.h>
#include <hip/hip_bf16.h>

// ---------------------------------------------------------------------------
// ChebNet forward for MI455X (gfx1250).
//   - CSR-by-dst built once per call -> gather-based propagation (no per-step
//     scatter atomics), all working sets L2-resident.
//   - Fused Chebyshev step kernel: per 16-node tile, compute
//     T_next = 2*prop(T_cur) - T_prev, stash tile in LDS, then accumulate
//     OUT += T_next @ W[k] with V_WMMA_F32_16X16X4_F32 (native fp32 WMMA).
// ---------------------------------------------------------------------------

typedef float v2f __attribute__((ext_vector_type(2)));
typedef float v8f __attribute__((ext_vector_type(8)));

#define NPAD     64   // padded feature width (all H/T buffers are N x 64)
#define TSTRIDE  68   // LDS tile row stride in floats (bank-conflict pad)
#define WSTRIDE  64   // LDS weight row stride

// ---------------------------- preprocessing --------------------------------

__global__ void k_zero_f(float* __restrict__ p, int n) {
    int i = blockIdx.x * blockDim.x + threadIdx.x;
    if (i < n) p[i] = 0.0f;
}

__global__ void k_zero_i(int* __restrict__ p, int n) {
    int i = blockIdx.x * blockDim.x + threadIdx.x;
    if (i < n) p[i] = 0;
}

// deg = segment_sum(ew, src); indeg = histogram(dst)
__global__ void k_deg_count(const int* __restrict__ src, const int* __restrict__ dst,
                            const float* __restrict__ ew,
                            float* __restrict__ deg, int* __restrict__ indeg, int E) {
    int e = blockIdx.x * blockDim.x + threadIdx.x;
    if (e < E) {
        atomicAdd(&deg[src[e]], ew[e]);
        atomicAdd(&indeg[dst[e]], 1);
    }
}

__global__ void k_dis(const float* __restrict__ deg, float* __restrict__ dis, int n) {
    int i = blockIdx.x * blockDim.x + threadIdx.x;
    if (i < n) {
        float d = deg[i];
        dis[i] = (d > 0.0f) ? __frsqrt_rn(d) : 0.0f;
    }
}

// single-block exclusive scan of indeg -> row_ptr (and cursor copy)
__global__ __launch_bounds__(1024) void k_scan(const int* __restrict__ indeg,
                                               int* __restrict__ row_ptr,
                                               int* __restrict__ cursor, int n) {
    __shared__ int buf[1024];
    __shared__ int carry_s;
    const int tid = threadIdx.x;
    if (tid == 0) carry_s = 0;
    __syncthreads();
    for (int base = 0; base < n; base += 1024) {
        int i = base + tid;
        int v = (i < n) ? indeg[i] : 0;
        buf[tid] = v;
        __syncthreads();
        for (int off = 1; off < 1024; off <<= 1) {
            int t = (tid >= off) ? buf[tid - off] : 0;
            __syncthreads();
            buf[tid] += t;
            __syncthreads();
        }
        int carry = carry_s;
        int excl  = carry + buf[tid] - v;
        if (i < n) { row_ptr[i] = excl; cursor[i] = excl; }
        int total = buf[1023];
        __syncthreads();
        if (tid == 0) carry_s = carry + total;
        __syncthreads();
    }
    if (tid == 0) row_ptr[n] = carry_s;
}

// wn = -dis[src]*ew*dis[dst]; scatter (src, wn) into CSR-by-dst slots
__global__ void k_fill(const int* __restrict__ src, const int* __restrict__ dst,
                       const float* __restrict__ ew, const float* __restrict__ dis,
                       int* __restrict__ cursor,
                       int* __restrict__ col, float* __restrict__ val, int E) {
    int e = blockIdx.x * blockDim.x + threadIdx.x;
    if (e < E) {
        int s = src[e], d = dst[e];
        float w = -dis[s] * ew[e] * dis[d];
        int p = atomicAdd(&cursor[d], 1);
        col[p] = s;
        val[p] = w;
    }
}

// H0[i, 0:64] = {x[i,0], x[i,1], 0...}
__global__ void k_pack_x(const float* __restrict__ x, float* __restrict__ H, int n) {
    int idx = blockIdx.x * blockDim.x + threadIdx.x;
    if (idx < n * NPAD) {
        int i = idx >> 6, f = idx & 63;
        H[idx] = (f < 2) ? x[i * 2 + f] : 0.0f;
    }
}

// ---------------------------- fused Chebyshev step -------------------------
// mode 0: tile = Hin rows (T0 term, no propagation, no T write)
// mode 1: T_next = prop(T_cur);                store T_next, tile = T_next
// mode 2: T_next = 2*prop(T_cur) - T_prev;     store T_next, tile = T_next
// then: OUT(16x64 tile) += tile(16x64) @ W[k](64x64, zero-padded, in LDS)
__global__ __launch_bounds__(256) void k_cheb_step(
    const float* __restrict__ Tprev, const float* __restrict__ Tcur,
    float* __restrict__ Tnext, float* __restrict__ OUT,
    const float* __restrict__ Wk,
    const int* __restrict__ row_ptr, const int* __restrict__ colidx,
    const float* __restrict__ val,
    int n_nodes, int Fin, int Fout, int mode)
{
    __shared__ float wlds[64 * WSTRIDE];           // 16 KB: padded W[k]
    __shared__ float tlds[8 * 16 * TSTRIDE];       // 34 KB: 8 wave tiles

    const int tid = threadIdx.x;

    // stage W[k] (Fin x Fout row-major) zero-padded to 64x64
    for (int idx = tid; idx < 64 * 64; idx += 256) {
        int r = idx >> 6, c = idx & 63;
        wlds[r * WSTRIDE + c] = (r < Fin && c < Fout) ? Wk[r * Fout + c] : 0.0f;
    }

    const int wave  = tid >> 5;
    const int lane  = tid & 31;
    const int mrow  = lane & 15;          // node row within 16-row tile
    const int half  = lane >> 4;          // which 32-feature half this lane owns
    const int mbase = (blockIdx.x * 8 + wave) * 16;
    const int node  = mbase + mrow;
    const int fbase = half * 32;
    const int fin4  = (Fin + 3) & ~3;     // gather bound (round up to float4)
    const bool valid = node < n_nodes;

    float* tile = &tlds[wave * 16 * TSTRIDE];

    float4 acc[8];
#pragma unroll
    for (int j = 0; j < 8; ++j) acc[j] = make_float4(0.f, 0.f, 0.f, 0.f);

    if (mode == 0) {
        if (valid) {
            const float4* srcp = (const float4*)(Tcur + (size_t)node * NPAD + fbase);
#pragma unroll
            for (int j = 0; j < 8; ++j) acc[j] = srcp[j];
        }
    } else {
        if (valid) {
            int e0 = row_ptr[node], e1 = row_ptr[node + 1];
            for (int e = e0; e < e1; ++e) {
                int   c = colidx[e];
                float w = val[e];
                const float4* srcp = (const float4*)(Tcur + (size_t)c * NPAD + fbase);
#pragma unroll
                for (int j = 0; j < 8; ++j) {
                    if (fbase + 4 * j < fin4) {        // skip all-zero pad chunks
                        float4 v = srcp[j];
                        acc[j].x += w * v.x; acc[j].y += w * v.y;
                        acc[j].z += w * v.z; acc[j].w += w * v.w;
                    }
                }
            }
            if (mode == 2) {
                const float4* tp = (const float4*)(Tprev + (size_t)node * NPAD + fbase);
#pragma unroll
                for (int j = 0; j < 8; ++j) {
                    float4 p = tp[j];
                    acc[j].x = 2.f * acc[j].x - p.x; acc[j].y = 2.f * acc[j].y - p.y;
                    acc[j].z = 2.f * acc[j].z - p.z; acc[j].w = 2.f * acc[j].w - p.w;
                }
            }
            float4* dstp = (float4*)(Tnext + (size_t)node * NPAD + fbase);
#pragma unroll
            for (int j = 0; j < 8; ++j) dstp[j] = acc[j];
        }
    }

    // tile rows for out-of-range nodes stay zero (acc never touched)
    {
        float4* trow = (float4*)(tile + mrow * TSTRIDE + fbase);
#pragma unroll
        for (int j = 0; j < 8; ++j) trow[j] = acc[j];
    }
    __syncthreads();

    // ---------------- WMMA: OUT[16x16 tiles] += tile @ wlds ----------------
    const int ntiles = (Fout + 15) >> 4;
    const int ksteps = (Fin + 3) >> 2;
    const int nlane  = lane & 15;         // N index within 16-col tile
    const int rowoff = half * 8;          // C layout: lanes 16-31 hold M = v+8

    for (int nt = 0; nt < ntiles; ++nt) {
        v8f c;
#pragma unroll
        for (int v = 0; v < 8; ++v) {
            int m = mbase + rowoff + v;
            c[v] = (m < n_nodes) ? OUT[(size_t)m * NPAD + nt * 16 + nlane] : 0.0f;
        }
        for (int kk = 0; kk < ksteps; ++kk) {
            int k0 = kk * 4;
            // A frag (16x4 f32): lane L<16 -> T[L][k0+0..1]; lane L>=16 -> T[L-16][k0+2..3]
            const float* ap = tile + mrow * TSTRIDE + k0 + 2 * half;
            v2f a; a.x = ap[0]; a.y = ap[1];
            // B frag (4x16 f32): VGPR0 rows {k0, k0+2}, VGPR1 rows {k0+1, k0+3}
            const float* bp = wlds + (k0 + 2 * half) * WSTRIDE + nt * 16 + nlane;
            v2f b; b.x = bp[0]; b.y = bp[WSTRIDE];
            c = __builtin_amdgcn_wmma_f32_16x16x4_f32(
                    false, a, false, b, (short)0, c, false, false);
        }
#pragma unroll
        for (int v = 0; v < 8; ++v) {
            int m = mbase + rowoff + v;
            if (m < n_nodes) OUT[(size_t)m * NPAD + nt * 16 + nlane] = c[v];
        }
    }
}

// ---------------------------- epilogues ------------------------------------

__global__ void k_silu(const float* __restrict__ OUT, const float* __restrict__ b,
                       float* __restrict__ H, int n, int Fout) {
    int idx = blockIdx.x * blockDim.x + threadIdx.x;
    if (idx < n * NPAD) {
        int f = idx & 63;
        float r = 0.0f;
        if (f < Fout) {
            float h = OUT[idx] + b[f];
            r = h / (1.0f + __expf(-h));
        }
        H[idx] = r;
    }
}

// final K=1 Cheb layer (30 -> 1) + sigmoid
__global__ void k_final(const float* __restrict__ H, const float* __restrict__ w4,
                        float* __restrict__ out, int n) {
    int i = blockIdx.x * blockDim.x + threadIdx.x;
    if (i < n) {
        float s = 0.0f;
#pragma unroll
        for (int f = 0; f < 30; ++f) s += H[(size_t)i * NPAD + f] * w4[f];
        out[i] = 1.0f / (1.0f + __expf(-s));
    }
}

// ---------------------------- host orchestration ---------------------------

extern "C" void kernel_launch(void* const* d_in, const int* in_sizes, int n_in,
                              void* d_out, int out_size, void* d_ws, size_t ws_size,
                              hipStream_t stream) {
    const float* x  = (const float*)d_in[0];
    const int*   ei = (const int*)  d_in[1];   // (2, E) int32
    const float* ew = (const float*)d_in[2];
    const float* W1 = (const float*)d_in[3];  const float* b1 = (const float*)d_in[4];
    const float* W2 = (const float*)d_in[5];  const float* b2 = (const float*)d_in[6];
    const float* W3 = (const float*)d_in[7];  const float* b3 = (const float*)d_in[8];
    const float* W4 = (const float*)d_in[9];

    const int N = in_sizes[0] / 2;
    const int E = in_sizes[2];
    const int* src = ei;
    const int* dst = ei + E;

    // carve scratch
    char* p = (char*)d_ws;
    auto carve = [&](size_t bytes) -> void* {
        void* r = (void*)p;
        p += (bytes + 255) & ~(size_t)255;
        return r;
    };
    float* deg     = (float*)carve((size_t)N * 4);
    float* dis     = (float*)carve((size_t)N * 4);
    int*   indeg   = (int*)  carve((size_t)N * 4);
    int*   row_ptr = (int*)  carve((size_t)(N + 1) * 4);
    int*   cursor  = (int*)  carve((size_t)N * 4);
    int*   col     = (int*)  carve((size_t)E * 4);
    float* val     = (float*)carve((size_t)E * 4);
    float* H0      = (float*)carve((size_t)N * NPAD * 4);
    float* H1      = (float*)carve((size_t)N * NPAD * 4);
    float* TA      = (float*)carve((size_t)N * NPAD * 4);
    float* TB      = (float*)carve((size_t)N * NPAD * 4);
    float* OUT     = (float*)carve((size_t)N * NPAD * 4);
    (void)ws_size; (void)n_in;

    const int TB_ = 256;
    const int gN   = (N + TB_ - 1) / TB_;
    const int gE   = (E + TB_ - 1) / TB_;
    const int gNF  = (N * NPAD + TB_ - 1) / TB_;
    const int gCheb = (N + 127) / 128;      // 8 waves x 16 nodes per block

    // ---- preprocessing: normalization + CSR-by-dst ----
    k_zero_f<<<gN, TB_, 0, stream>>>(deg, N);
    k_zero_i<<<gN, TB_, 0, stream>>>(indeg, N);
    k_deg_count<<<gE, TB_, 0, stream>>>(src, dst, ew, deg, indeg, E);
    k_dis<<<gN, TB_, 0, stream>>>(deg, dis, N);
    k_scan<<<1, 1024, 0, stream>>>(indeg, row_ptr, cursor, N);
    k_fill<<<gE, TB_, 0, stream>>>(src, dst, ew, dis, cursor, col, val, E);
    k_pack_x<<<gNF, TB_, 0, stream>>>(x, H0, N);

    auto run_layer = [&](const float* W, const float* b, int K, int Fin, int Fout,
                         float* Hin, float* Hout) {
        k_zero_f<<<gNF, TB_, 0, stream>>>(OUT, N * NPAD);
        // k = 0: OUT += Hin @ W[0]
        k_cheb_step<<<gCheb, 256, 0, stream>>>(nullptr, Hin, nullptr, OUT, W,
                                               row_ptr, col, val, N, Fin, Fout, 0);
        if (K > 1) {
            // k = 1: T1 = L_hat @ Hin; OUT += T1 @ W[1]
            k_cheb_step<<<gCheb, 256, 0, stream>>>(nullptr, Hin, TA, OUT,
                                                   W + (size_t)Fin * Fout,
                                                   row_ptr, col, val, N, Fin, Fout, 1);
            float* tp = Hin; float* tc = TA; float* tn = TB;
            for (int k = 2; k < K; ++k) {
                k_cheb_step<<<gCheb, 256, 0, stream>>>(tp, tc, tn, OUT,
                                                       W + (size_t)k * Fin * Fout,
                                                       row_ptr, col, val, N, Fin, Fout, 2);
                float* t = tp; tp = tc; tc = tn; tn = t;
            }
        }
        k_silu<<<gNF, TB_, 0, stream>>>(OUT, b, Hout, N, Fout);
    };

    run_layer(W1, b1, 200,  2, 60, H0, H1);
    run_layer(W2, b2, 200, 60, 60, H1, H0);
    run_layer(W3, b3,  20, 60, 30, H0, H1);
    k_final<<<gN, TB_, 0, stream>>>(H1, W4, (float*)d_out, N);
    (void)out_size;
}